// BahdanauAttention_49555332662169
// MI455X (gfx1250) — compile-verified
//
#include <hip/hip_runtime.h>

// ---------------------------------------------------------------------------
// Bahdanau additive attention (HFTA-fused), MI455X / gfx1250.
//   B=2, b=16, t_q=64, t_k=128, d = qs = ks = nu = 512.
// Stage 1: pq = Q @ Wq^T, pk = K @ Wk^T        (fp32 WMMA 16x16x4)
// Stage 2: scores = softmax_k( sum_n tanh(pq+pk)*v )   (v_tanh_f32 TRANS)
// Stage 3: context = scores @ keys             (fp32 WMMA 16x16x4)
// ---------------------------------------------------------------------------

typedef __attribute__((ext_vector_type(2))) float v2f;
typedef __attribute__((ext_vector_type(4))) float v4f;
typedef __attribute__((ext_vector_type(8))) float v8f;

#define D       512
#define TQ      64
#define TK      128
#define SLICES  32          // B*b
#define PQ_ROWS 2048        // SLICES*TQ
#define PK_ROWS 4096        // SLICES*TK

__device__ __forceinline__ v8f wmma4(v2f a, v2f b, v8f c) {
    // V_WMMA_F32_16X16X4_F32: D(16x16 f32) = A(16x4 f32) * B(4x16 f32) + C
    return __builtin_amdgcn_wmma_f32_16x16x4_f32(
        /*neg_a=*/false, a, /*neg_b=*/false, b,
        /*c_mod=*/(short)0, c, /*reuse_a=*/false, /*reuse_b=*/false);
}

// Four hardware tanhs per asm block: each trans op is the required "one
// independent instruction" covering the previous one's RAW hazard; a single
// v_nop covers the last.  Early-clobber outputs so results never alias the
// still-to-be-read inputs.
__device__ __forceinline__ float tanh4_dot(v4f x, v4f w, float acc) {
    float t0, t1, t2, t3;
    asm("v_tanh_f32 %0, %4\n\t"
        "v_tanh_f32 %1, %5\n\t"
        "v_tanh_f32 %2, %6\n\t"
        "v_tanh_f32 %3, %7\n\t"
        "v_nop"
        : "=&v"(t0), "=&v"(t1), "=&v"(t2), "=&v"(t3)
        : "v"(x.x), "v"(x.y), "v"(x.z), "v"(x.w));
    acc = fmaf(t0, w.x, acc);
    acc = fmaf(t1, w.y, acc);
    acc = fmaf(t2, w.z, acc);
    acc = fmaf(t3, w.w, acc);
    return acc;
}

// ---------------------------------------------------------------------------
// Kernel 1: fused projections.  One wave -> one 16(M) x 64(N) output strip.
// A-frag layout (ISA 32-bit A 16x4): element (v, half) has K = v + 2*half,
//   M = ln16 -> contiguous per-lane pair at col k + 2*half.
// B-frag (4x16, row-striped like C): element (v, half) has K = v + 2*half,
//   N = ln16.  B[k][n] = W[n][k] (W row-major (out,in)) -> contiguous pair.
// ---------------------------------------------------------------------------
__global__ void __launch_bounds__(256)
proj_kernel(const float* __restrict__ query, const float* __restrict__ keys,
            const float* __restrict__ Wq,    const float* __restrict__ Wk,
            float* __restrict__ pq,          float* __restrict__ pk) {
    const int gwave = (blockIdx.x * 256 + (int)threadIdx.x) >> 5;   // 0..3071
    const int lane  = threadIdx.x & 31;
    const int strip = gwave & 7;          // N strip of 64
    const int mtile = gwave >> 3;         // 0..383  (128 pq tiles + 256 pk)

    const float *A, *W;
    float *O;
    if (mtile < PQ_ROWS / 16) {
        const int r0 = mtile * 16;                  // row over (B,b,q)
        A = query + (size_t)r0 * D;
        W = Wq + (size_t)(r0 / (16 * TQ)) * D * D;  // r0/1024 -> B index
        O = pq + (size_t)r0 * D;
    } else {
        const int r0 = (mtile - PQ_ROWS / 16) * 16; // row over (B,b,k)
        A = keys + (size_t)r0 * D;
        W = Wk + (size_t)(r0 / (16 * TK)) * D * D;  // r0/2048 -> B index
        O = pk + (size_t)r0 * D;
    }

    const int half = lane >> 4;
    const int ln16 = lane & 15;
    const int n0   = strip * 64;

    const float* aP  = A + (size_t)ln16 * D + 2 * half;
    const float* bP0 = W + (size_t)(n0 +  0 + ln16) * D + 2 * half;
    const float* bP1 = W + (size_t)(n0 + 16 + ln16) * D + 2 * half;
    const float* bP2 = W + (size_t)(n0 + 32 + ln16) * D + 2 * half;
    const float* bP3 = W + (size_t)(n0 + 48 + ln16) * D + 2 * half;

    v8f c0 = {}, c1 = {}, c2 = {}, c3 = {};
    #pragma unroll 4
    for (int k = 0; k < D; k += 4) {
        v2f a  = *(const v2f*)(aP  + k);
        v2f b0 = *(const v2f*)(bP0 + k);
        v2f b1 = *(const v2f*)(bP1 + k);
        v2f b2 = *(const v2f*)(bP2 + k);
        v2f b3 = *(const v2f*)(bP3 + k);
        c0 = wmma4(a, b0, c0);
        c1 = wmma4(a, b1, c1);
        c2 = wmma4(a, b2, c2);
        c3 = wmma4(a, b3, c3);
    }
    // C/D layout: element (vgpr r, lane): M = r + 8*half, N = ln16.
    float* oP = O + (size_t)(8 * half) * D + n0 + ln16;
    #pragma unroll
    for (int r = 0; r < 8; ++r) {
        oP[(size_t)r * D +  0] = c0[r];
        oP[(size_t)r * D + 16] = c1[r];
        oP[(size_t)r * D + 32] = c2[r];
        oP[(size_t)r * D + 48] = c3[r];
    }
}

// ---------------------------------------------------------------------------
// Kernel 2: scores + softmax.  One block per (B,b,q) row.
// Lane owns n in {4*lane + c + 128*j : c<4, j<4}: pq-row and v values for
// those 16 n live in registers for the whole k-loop (loaded once); pk is read
// as per-lane float4 (global_load_b128, coalesced across the wave).
// ---------------------------------------------------------------------------
__global__ void __launch_bounds__(256)
score_softmax_kernel(const float* __restrict__ pq, const float* __restrict__ pk,
                     const float* __restrict__ vatt, float* __restrict__ scores_out) {
    __shared__ float s_sc[TK];

    const int row   = blockIdx.x;        // 0..2047 over (B,b,q)
    const int slice = row / TQ;          // 0..31
    const int tid   = threadIdx.x;       // 0..255
    const int wave  = tid >> 5;
    const int lane  = tid & 31;
    const int nb    = lane * 4;          // 16B-aligned per-lane base

    const float* pqrow  = pq + (size_t)row * D + nb;
    const float* pkbase = pk + (size_t)slice * TK * D + nb;

    const v4f q0 = *(const v4f*)(pqrow);
    const v4f q1 = *(const v4f*)(pqrow + 128);
    const v4f q2 = *(const v4f*)(pqrow + 256);
    const v4f q3 = *(const v4f*)(pqrow + 384);
    const v4f w0 = *(const v4f*)(vatt + nb);
    const v4f w1 = *(const v4f*)(vatt + nb + 128);
    const v4f w2 = *(const v4f*)(vatt + nb + 256);
    const v4f w3 = *(const v4f*)(vatt + nb + 384);

    // 8 waves x 16 k-iterations cover all 128 keys.
    #pragma unroll 2
    for (int kk = 0; kk < TK / 8; ++kk) {
        const int k = kk * 8 + wave;
        const float* pkr = pkbase + (size_t)k * D;
        const v4f p0 = *(const v4f*)(pkr);
        const v4f p1 = *(const v4f*)(pkr + 128);
        const v4f p2 = *(const v4f*)(pkr + 256);
        const v4f p3 = *(const v4f*)(pkr + 384);
        float partial = 0.f;
        partial = tanh4_dot(q0 + p0, w0, partial);
        partial = tanh4_dot(q1 + p1, w1, partial);
        partial = tanh4_dot(q2 + p2, w2, partial);
        partial = tanh4_dot(q3 + p3, w3, partial);
        #pragma unroll
        for (int off = 16; off > 0; off >>= 1)
            partial += __shfl_xor(partial, off, 32);
        if (lane == 0) s_sc[k] = partial;
    }
    __syncthreads();

    // Softmax over 128 scores on wave 0 (4 values per lane).
    if (wave == 0) {
        float s0 = s_sc[lane], s1 = s_sc[lane + 32];
        float s2 = s_sc[lane + 64], s3 = s_sc[lane + 96];
        float m = fmaxf(fmaxf(s0, s1), fmaxf(s2, s3));
        #pragma unroll
        for (int off = 16; off > 0; off >>= 1)
            m = fmaxf(m, __shfl_xor(m, off, 32));
        float e0 = __expf(s0 - m), e1 = __expf(s1 - m);
        float e2 = __expf(s2 - m), e3 = __expf(s3 - m);
        float sum = e0 + e1 + e2 + e3;
        #pragma unroll
        for (int off = 16; off > 0; off >>= 1)
            sum += __shfl_xor(sum, off, 32);
        const float inv = 1.f / sum;
        float* out = scores_out + (size_t)row * TK;
        out[lane]      = e0 * inv;
        out[lane + 32] = e1 * inv;
        out[lane + 64] = e2 * inv;
        out[lane + 96] = e3 * inv;
    }
}

// ---------------------------------------------------------------------------
// Kernel 3: context = P(64x128) @ keys(128x512) per slice, fp32 WMMA.
// keys is row-major K x N here: b.x = keys[(k+2h)*D + n], b.y = next k-row.
// ---------------------------------------------------------------------------
__global__ void __launch_bounds__(256)
context_kernel(const float* __restrict__ probs, const float* __restrict__ keys,
               float* __restrict__ ctx) {
    const int gwave = (blockIdx.x * 256 + (int)threadIdx.x) >> 5;  // 0..1023
    const int lane  = threadIdx.x & 31;
    const int strip = gwave & 7;          // N strip of 64
    const int mt    = (gwave >> 3) & 3;   // M tile (64/16)
    const int s     = gwave >> 5;         // slice 0..31

    const int half = lane >> 4;
    const int ln16 = lane & 15;
    const int n0   = strip * 64;

    const float* A = probs + (size_t)s * TQ * TK + (size_t)mt * 16 * TK;
    const float* B = keys  + (size_t)s * TK * D;
    float*       O = ctx   + (size_t)s * TQ * D  + (size_t)mt * 16 * D;

    const float* aP = A + (size_t)ln16 * TK + 2 * half;
    const float* bP = B + (size_t)(2 * half) * D + n0 + ln16;

    v8f c0 = {}, c1 = {}, c2 = {}, c3 = {};
    #pragma unroll 4
    for (int k = 0; k < TK; k += 4) {
        v2f a = *(const v2f*)(aP + k);
        const float* bk = bP + (size_t)k * D;
        v2f b0, b1, b2, b3;
        b0.x = bk[ 0]; b0.y = bk[D +  0];
        b1.x = bk[16]; b1.y = bk[D + 16];
        b2.x = bk[32]; b2.y = bk[D + 32];
        b3.x = bk[48]; b3.y = bk[D + 48];
        c0 = wmma4(a, b0, c0);
        c1 = wmma4(a, b1, c1);
        c2 = wmma4(a, b2, c2);
        c3 = wmma4(a, b3, c3);
    }
    float* oP = O + (size_t)(8 * half) * D + n0 + ln16;
    #pragma unroll
    for (int r = 0; r < 8; ++r) {
        oP[(size_t)r * D +  0] = c0[r];
        oP[(size_t)r * D + 16] = c1[r];
        oP[(size_t)r * D + 32] = c2[r];
        oP[(size_t)r * D + 48] = c3[r];
    }
}

// ---------------------------------------------------------------------------
extern "C" void kernel_launch(void* const* d_in, const int* in_sizes, int n_in,
                              void* d_out, int out_size, void* d_ws, size_t ws_size,
                              hipStream_t stream) {
    const float* query = (const float*)d_in[0];  // (2,16,64,512)
    const float* keys  = (const float*)d_in[1];  // (2,16,128,512)
    const float* Wq    = (const float*)d_in[2];  // (2,512,512)
    const float* Wk    = (const float*)d_in[3];  // (2,512,512)
    const float* vatt  = (const float*)d_in[4];  // (512,)

    float* ctx    = (float*)d_out;                          // 2*16*64*512
    float* scores = ctx + (size_t)SLICES * TQ * D;          // 2*16*64*128

    float* pq = (float*)d_ws;                               // 2048*512 f32 (4 MB)
    float* pk = pq + (size_t)PQ_ROWS * D;                   // 4096*512 f32 (8 MB)

    // Stage 1: 3072 wave-tasks, 8 waves/block -> 384 blocks.
    proj_kernel<<<384, 256, 0, stream>>>(query, keys, Wq, Wk, pq, pk);
    // Stage 2: one block per (B,b,q) row.
    score_softmax_kernel<<<PQ_ROWS, 256, 0, stream>>>(pq, pk, vatt, scores);
    // Stage 3: 1024 wave-tasks -> 128 blocks.
    context_kernel<<<128, 256, 0, stream>>>(scores, keys, ctx);
}